// GraphLaplacianSE_19112604467701
// MI455X (gfx1250) — compile-verified
//
#include <hip/hip_runtime.h>

// GraphLaplacian: out[i] = sum over edges (i->j) of (pos[i] - pos[j])
// == deg(i)*pos[i] - sum_j pos[j]  (matches reference, incl. self-edges)

// clang ext vector so __builtin_nontemporal_load accepts it (HIP's int4 is a class).
typedef int v4i __attribute__((ext_vector_type(4)));

// --- Compile-only probe (never launched): shows the exact lowering of the
// relaxed/agent FP32 atomic add at the very top of the disassembly snippet.
// Expect a single native global_atomic_add_f32 (no CAS loop).
__global__ void gl_atomic_probe_kernel(float* p, float v) {
    __hip_atomic_fetch_add(p, v, __ATOMIC_RELAXED, __HIP_MEMORY_SCOPE_AGENT);
}

// Flat hot kernel: one quad of edges per thread.
__global__ __launch_bounds__(256) void gl_edge_scatter4_kernel(
    const v4i* __restrict__ src4, const v4i* __restrict__ dst4,
    const float* __restrict__ pos, float* __restrict__ out, int n_quads) {
    int t = blockIdx.x * blockDim.x + threadIdx.x;
    if (t >= n_quads) return;

    // Warm L2 one resident-window (~2 MB) ahead; branchless clamp keeps the
    // prefetch address valid.
    int qp = t + (1 << 17);
    if (qp >= n_quads) qp = t;
    __builtin_prefetch((const void*)&src4[qp], 0, 0);
    __builtin_prefetch((const void*)&dst4[qp], 0, 0);

    // Non-temporal 128-bit streaming loads of the edge list.
    v4i s = __builtin_nontemporal_load(&src4[t]);
    v4i d = __builtin_nontemporal_load(&dst4[t]);
#pragma unroll
    for (int k = 0; k < 4; ++k) {
        int si = s[k];
        int di = d[k];
        // pos is 1.2 MB -> L2/WGP$ resident; regular temporal loads.
        float cx = pos[3 * si + 0] - pos[3 * di + 0];
        float cy = pos[3 * si + 1] - pos[3 * di + 1];
        float cz = pos[3 * si + 2] - pos[3 * di + 2];
        // Native global_atomic_add_f32 (no return).
        __hip_atomic_fetch_add(&out[3 * si + 0], cx, __ATOMIC_RELAXED,
                               __HIP_MEMORY_SCOPE_AGENT);
        __hip_atomic_fetch_add(&out[3 * si + 1], cy, __ATOMIC_RELAXED,
                               __HIP_MEMORY_SCOPE_AGENT);
        __hip_atomic_fetch_add(&out[3 * si + 2], cz, __ATOMIC_RELAXED,
                               __HIP_MEMORY_SCOPE_AGENT);
    }
}

__global__ __launch_bounds__(256) void gl_zero_kernel(float* __restrict__ out, int n) {
    int i = blockIdx.x * blockDim.x + threadIdx.x;
    if (i < n) out[i] = 0.0f;
}

// Scalar tail for edge counts not divisible by 4 (not hit for 6.4M, kept for safety).
__global__ __launch_bounds__(256) void gl_edge_scatter1_kernel(
    const int* __restrict__ src, const int* __restrict__ dst,
    const float* __restrict__ pos, float* __restrict__ out,
    int first_edge, int n_edges) {
    int e = first_edge + blockIdx.x * blockDim.x + threadIdx.x;
    if (e >= n_edges) return;
    int si = src[e];
    int di = dst[e];
    float cx = pos[3 * si + 0] - pos[3 * di + 0];
    float cy = pos[3 * si + 1] - pos[3 * di + 1];
    float cz = pos[3 * si + 2] - pos[3 * di + 2];
    __hip_atomic_fetch_add(&out[3 * si + 0], cx, __ATOMIC_RELAXED, __HIP_MEMORY_SCOPE_AGENT);
    __hip_atomic_fetch_add(&out[3 * si + 1], cy, __ATOMIC_RELAXED, __HIP_MEMORY_SCOPE_AGENT);
    __hip_atomic_fetch_add(&out[3 * si + 2], cz, __ATOMIC_RELAXED, __HIP_MEMORY_SCOPE_AGENT);
}

extern "C" void kernel_launch(void* const* d_in, const int* in_sizes, int n_in,
                              void* d_out, int out_size, void* d_ws, size_t ws_size,
                              hipStream_t stream) {
    const int*   edge_index = (const int*)d_in[0];   // [2, E] flat: row 0 = src, row 1 = dst
    const float* pos        = (const float*)d_in[1]; // [N, 3]
    float*       out        = (float*)d_out;         // [N, 3]

    int n_edges = in_sizes[0] / 2;
    const int* src = edge_index;
    const int* dst = edge_index + n_edges;

    // 1) Zero the accumulator (d_out is poisoned by the harness).
    {
        int n = out_size;
        int blocks = (n + 255) / 256;
        gl_zero_kernel<<<blocks, 256, 0, stream>>>(out, n);
    }

    // 2) Scatter-add edge contributions: one quad per thread.
    int n_quads = n_edges >> 2;
    if (n_quads > 0) {
        int blocks = (n_quads + 255) / 256;
        gl_edge_scatter4_kernel<<<blocks, 256, 0, stream>>>(
            (const v4i*)src, (const v4i*)dst, pos, out, n_quads);
    }
    int tail_start = n_quads << 2;
    int tail = n_edges - tail_start;
    if (tail > 0) {
        int blocks = (tail + 255) / 256;
        gl_edge_scatter1_kernel<<<blocks, 256, 0, stream>>>(
            src, dst, pos, out, tail_start, n_edges);
    }
}